// EM15Temp_25829933318538
// MI455X (gfx1250) — compile-verified
//
#include <hip/hip_runtime.h>
#include <stdint.h>

// entmax-1.5 over last dim (d=64) of (1024,1024,64) fp32.
// One thread per row; Batcher odd-even mergesort (543 comparators) of 64
// values fully in VGPRs. Rows staged into LDS by the CDNA5 Tensor Data Mover
// with per-row padding (272B stride) so per-lane LDS reads are conflict-free.

#define D               64
#define ROW_BYTES       256            // 64 * 4B
#define LDS_ROW_STRIDE  272            // 256 + 16B TDM pad -> conflict-free LDS reads
#define ROWS_PER_BLOCK  128
#define THREADS         128            // 4 waves, 32 rows per wave

typedef unsigned int v4u __attribute__((ext_vector_type(4)));
typedef int          v8i __attribute__((ext_vector_type(8)));
typedef int          v4i __attribute__((ext_vector_type(4)));

// ---- Batcher odd-even mergesort, descending, compile-time network ----
__device__ __forceinline__ void ce_desc(float* v, int a, int b) {
    float x = v[a], y = v[b];
    v[a] = fmaxf(x, y);                // larger value to lower index
    v[b] = fminf(x, y);
}

template<int LO, int N, int R>
__device__ __forceinline__ void oemerge(float* v) {
    constexpr int M = R * 2;
    if constexpr (M < N) {
        oemerge<LO, N, M>(v);          // even subsequence
        oemerge<LO + R, N, M>(v);      // odd subsequence
        #pragma unroll
        for (int i = LO + R; i + R < LO + N; i += M)
            ce_desc(v, i, i + R);
    } else {
        ce_desc(v, LO, LO + R);
    }
}

template<int LO, int N>
__device__ __forceinline__ void oesort(float* v) {
    if constexpr (N > 1) {
        constexpr int M = N / 2;
        oesort<LO, M>(v);
        oesort<LO + M, M>(v);
        oemerge<LO, N, 1>(v);
    }
}

__global__ __launch_bounds__(THREADS) void entmax15_kernel(
    const float* __restrict__ in, float* __restrict__ out, int nrows)
{
    __shared__ __align__(16) unsigned char smem[ROWS_PER_BLOCK * LDS_ROW_STRIDE];

    const int tid  = threadIdx.x;
    const int wave = tid >> 5;
    const int row  = blockIdx.x * ROWS_PER_BLOCK + tid;

#if __has_builtin(__builtin_amdgcn_tensor_load_to_lds)
    // ---- TDM: each wave DMAs its 32 rows (a 64x32 fp32 2-D tile) into LDS ----
    {
        uint64_t gaddr = (uint64_t)(uintptr_t)in +
                         (uint64_t)(blockIdx.x * ROWS_PER_BLOCK + wave * 32) * ROW_BYTES;
        uint32_t laddr = (uint32_t)(uintptr_t)smem + (uint32_t)(wave * 32 * LDS_ROW_STRIDE);

        // D# group 0 (CDNA5 ISA)
        v4u g0;
        g0[0] = 1u;                                   // count=1 (valid), gather off
        g0[1] = laddr;                                // lds_addr (bytes)
        g0[2] = (uint32_t)gaddr;                      // global_addr[31:0]
        g0[3] = (uint32_t)((gaddr >> 32) & 0x01FFFFFFu) | (2u << 30); // addr[56:32], type=2

        // D# group 1 (§8.4)
        v8i g1;
        g1[0] = (int)((2u << 16)      // data_size = 4 bytes
                    | (1u << 20)      // pad_enable
                    | (5u << 22)      // pad_interval: 64 DWORDs (one row)
                    | (3u << 25));    // pad_amount: 4 DWORDs (16B) -> 272B row stride
        g1[1] = (int)(64u << 16);     // tensor_dim0[15:0] = 64 (bits 79:48 span d1/d2)
        g1[2] = 0;                    // tensor_dim0 hi = 0, tensor_dim1 lo16 = 0
        g1[3] = (int)(0x10u | (64u << 16)); // tensor_dim1 hi16 (1<<20 rows), tile_dim0 = 64
        g1[4] = 32;                   // tile_dim1 = 32 rows, tile_dim2 = 0
        g1[5] = 64;                   // tensor_dim0_stride = 64 elements (low 32)
        g1[6] = 0;                    // stride0 hi16 = 0, dim1_stride lo16 = 0 (unused, 2-D)
        g1[7] = 0;

        v4i g2 = {0, 0, 0, 0};        // 2-D tensor: groups 2/3 unused
        v4i g3 = {0, 0, 0, 0};
        v8i g4 = {0, 0, 0, 0, 0, 0, 0, 0}; // extra group (6-arg clang form)

        __builtin_amdgcn_tensor_load_to_lds(g0, g1, g2, g3, g4, 0);
        __builtin_amdgcn_s_wait_tensorcnt(0);   // s_wait_tensorcnt 0
        // Each wave reads only rows it loaded -> no workgroup barrier required.
    }
#else
    // Fallback: cooperative coalesced copy global -> padded LDS.
    {
        const float4* in4 = (const float4*)(in) + (size_t)blockIdx.x * (ROWS_PER_BLOCK * 16);
        #pragma unroll
        for (int it = 0; it < (ROWS_PER_BLOCK * 16) / THREADS; ++it) {
            int idx = it * THREADS + tid;
            int r = idx >> 4, c = idx & 15;
            *((float4*)(smem + r * LDS_ROW_STRIDE) + c) = in4[idx];
        }
        __syncthreads();
    }
#endif

    if (row >= nrows) return;

    const float* lrow = (const float*)(smem + tid * LDS_ROW_STRIDE);

    // ---- Load own row from LDS into registers ----
    float v[D];
    #pragma unroll
    for (int j = 0; j < D / 4; ++j) {
        float4 q = ((const float4*)lrow)[j];
        v[4 * j + 0] = q.x; v[4 * j + 1] = q.y;
        v[4 * j + 2] = q.z; v[4 * j + 3] = q.w;
    }

    // ---- Sort descending: Batcher odd-even merge network (543 CEs) ----
    oesort<0, D>(v);

    // ---- Prefix scan: tau_k = mean_k - sqrt(max((1 - ss_k)/rho, 0)); keep last
    //      k with tau_k <= xs_k (support is a prefix for entmax). ----
    const float m  = v[0];
    const float hm = 0.5f * m;
    float cs = 0.0f;          // sum of transformed sorted prefix
    float os = 1.0f;          // 1 - sum of squares
    float tau_star = 0.0f;
    #pragma unroll
    for (int k = 0; k < D; ++k) {
        float y = __builtin_fmaf(0.5f, v[k], -hm);     // transformed sorted value
        cs += y;
        os  = __builtin_fmaf(-y, y, os);               // 1 - ss
        float rinv  = 1.0f / (float)(k + 1);           // compile-time constant
        float mean  = cs * rinv;
        float inner = __builtin_fmaf(cs, mean, os);    // 1 - ss + cs^2/rho
        float delta = fmaxf(inner * rinv, 0.0f);
        float tau   = mean - __builtin_amdgcn_sqrtf(delta); // v_sqrt_f32 (trans pipe)
        tau_star    = (tau <= y) ? tau : tau_star;
    }

    // ---- Output: clip(0.5*(x - max) - tau*, 0)^2, re-reading raw x from LDS ----
    const float c = hm + tau_star;
    float4* orow = (float4*)(out + (size_t)row * D);
    #pragma unroll
    for (int j = 0; j < D / 4; ++j) {
        float4 q = ((const float4*)lrow)[j];
        float4 r;
        r.x = fmaxf(__builtin_fmaf(0.5f, q.x, -c), 0.0f); r.x *= r.x;
        r.y = fmaxf(__builtin_fmaf(0.5f, q.y, -c), 0.0f); r.y *= r.y;
        r.z = fmaxf(__builtin_fmaf(0.5f, q.z, -c), 0.0f); r.z *= r.z;
        r.w = fmaxf(__builtin_fmaf(0.5f, q.w, -c), 0.0f); r.w *= r.w;
        orow[j] = r;
    }
}

extern "C" void kernel_launch(void* const* d_in, const int* in_sizes, int n_in,
                              void* d_out, int out_size, void* d_ws, size_t ws_size,
                              hipStream_t stream) {
    const float* in  = (const float*)d_in[0];
    float*       out = (float*)d_out;
    const int nrows  = in_sizes[0] / D;                 // 1,048,576
    const int blocks = (nrows + ROWS_PER_BLOCK - 1) / ROWS_PER_BLOCK;
    hipLaunchKernelGGL(entmax15_kernel, dim3(blocks), dim3(THREADS), 0, stream,
                       in, out, nrows);
}